// TSumTDisc_54717883351108
// MI455X (gfx1250) — compile-verified
//
#include <hip/hip_runtime.h>

typedef __attribute__((ext_vector_type(2))) float v2f;
typedef __attribute__((ext_vector_type(8))) float v8f;

#define NUM_GRAPHS 512
#define N_NODES    262144
#define N_FTX      16
#define LRELU_SLOPE 0.01f

__device__ __forceinline__ float lrelu(float v) { return v > 0.f ? v : LRELU_SLOPE * v; }

__device__ __forceinline__ v8f v8f_zero() {
  v8f z;
#pragma unroll
  for (int i = 0; i < 8; ++i) z[i] = 0.f;
  return z;
}

// D = A(16x4,f32) * B(4x16,f32) + C(16x16,f32) -- exact fp32 matrix op on CDNA5
__device__ __forceinline__ v8f wmma4(v2f a, v2f b, v8f c) {
  return __builtin_amdgcn_wmma_f32_16x16x4_f32(false, a, false, b, (short)0, c, false, false);
}

// ---------------------------------------------------------------------------
// Fused per-node 3-layer MLP: [16 x NIN] -> 64 -> 64 -> 16 per wave-tile.
// Weights live in LDS in a K-paired layout so each B fragment is a single
// aligned ds_load_b64:  addr(k,col) = (k>>1)*2*NOUT + col*2 + (k&1).
// Inter-layer C->A layout conversion bounces through a private LDS tile
// (wave32, DS in-order per wave, explicit s_wait_dscnt).
// ---------------------------------------------------------------------------
template <int NIN, bool CONCAT, bool FINAL_LIN, bool RESID>
__global__ void __launch_bounds__(128)
node_ffn_kernel(const float* __restrict__ in,      // [N,16] node input (x or h)
                const float* __restrict__ xglob,   // [G,32] (CONCAT only)
                const int*   __restrict__ batch,   // [N]    (CONCAT only)
                const float* __restrict__ resid,   // [N,16] (RESID only)
                const float* __restrict__ gW1, const float* __restrict__ gB1,
                const float* __restrict__ gW2, const float* __restrict__ gB2,
                const float* __restrict__ gW3, const float* __restrict__ gB3,
                float* __restrict__ out)           // [N,16]
{
  __shared__ float s_w1[NIN * 64];    // K-paired, NOUT=64
  __shared__ float s_w2[64 * 64];     // K-paired, NOUT=64
  __shared__ float s_w3[64 * 16];     // K-paired, NOUT=16
  __shared__ float s_b1[64];
  __shared__ float s_b2[64];
  __shared__ float s_b3[16];
  __shared__ float s_bounce[4][16 * 64];

  const int tid = threadIdx.x;
  for (int i = tid; i < NIN * 64; i += 128) {
    const int k = i >> 6, c = i & 63;
    s_w1[((k >> 1) << 7) + (c << 1) + (k & 1)] = gW1[i];
  }
  for (int i = tid; i < 64 * 64; i += 128) {
    const int k = i >> 6, c = i & 63;
    s_w2[((k >> 1) << 7) + (c << 1) + (k & 1)] = gW2[i];
  }
  for (int i = tid; i < 64 * 16; i += 128) {
    const int k = i >> 4, c = i & 15;
    s_w3[((k >> 1) << 5) + (c << 1) + (k & 1)] = gW3[i];
  }
  if (tid < 64) { s_b1[tid] = gB1[tid]; s_b2[tid] = gB2[tid]; }
  if (tid < 16) s_b3[tid] = gB3[tid];
  __syncthreads();

  const int lane = tid & 31;
  const int wave = tid >> 5;
  float* bounce = &s_bounce[wave][0];
  const int n    = lane & 15;          // column (B/C) or row (A) within tile
  const int hi   = lane >> 4;          // half-wave id
  const int hi2  = hi << 1;            // K sub-offset for A/B frags (k0 even)
  const int mofs = hi << 3;            // M offset for C/D rows
  float* brow = bounce + (n << 6);     // this lane's A-row in the bounce tile

  // tile-invariant per-lane bias scalars
  float b1v[4], b2v[4];
#pragma unroll
  for (int t = 0; t < 4; ++t) { b1v[t] = s_b1[t * 16 + n]; b2v[t] = s_b2[t * 16 + n]; }
  const float b3v = s_b3[n];

  const int ntiles  = N_NODES / 16;
  const int wstride = gridDim.x * 4;
  for (int tile = blockIdx.x * 4 + wave; tile < ntiles; tile += wstride) {
    const int r0   = tile * 16;
    const int arow = r0 + n;                          // A-matrix row for this lane
    const float* inrow = in + (size_t)arow * 16;
    const float* xgrow = nullptr;
    if (CONCAT) xgrow = xglob + (size_t)batch[arow] * 32;

    // ---- layer 1: [16 x NIN] @ [NIN x 64] + b1, lrelu ----
    v8f c1[4];
#pragma unroll
    for (int t = 0; t < 4; ++t) c1[t] = v8f_zero();
#pragma unroll
    for (int kk = 0; kk < NIN / 4; ++kk) {
      const int k0 = kk * 4 + hi2;
      const int kp = k0 >> 1;                         // = kk*2 + hi
      v2f a;
      if (!CONCAT) {
        a = *(const v2f*)(inrow + k0);
      } else {
        a = (k0 < 16) ? *(const v2f*)(inrow + k0)
                      : *(const v2f*)(xgrow + (k0 - 16));
      }
#pragma unroll
      for (int t = 0; t < 4; ++t) {
        const v2f b = *(const v2f*)&s_w1[(kp << 7) + ((t * 16 + n) << 1)];
        c1[t] = wmma4(a, b, c1[t]);
      }
    }
#pragma unroll
    for (int t = 0; t < 4; ++t) {
#pragma unroll
      for (int r = 0; r < 8; ++r)
        bounce[(r + mofs) * 64 + t * 16 + n] = lrelu(c1[t][r] + b1v[t]);
    }
    asm volatile("s_wait_dscnt 0" ::: "memory");

    // ---- layer 2: [16 x 64] @ [64 x 64] + b2, lrelu ----
    v8f c2[4];
#pragma unroll
    for (int t = 0; t < 4; ++t) c2[t] = v8f_zero();
#pragma unroll
    for (int kk = 0; kk < 16; ++kk) {
      const int k0 = kk * 4 + hi2;
      const int kp = k0 >> 1;
      const v2f a = *(const v2f*)(brow + k0);
#pragma unroll
      for (int t = 0; t < 4; ++t) {
        const v2f b = *(const v2f*)&s_w2[(kp << 7) + ((t * 16 + n) << 1)];
        c2[t] = wmma4(a, b, c2[t]);
      }
    }
    asm volatile("s_wait_dscnt 0" ::: "memory");  // all reads of bounce done (in-order DS)
#pragma unroll
    for (int t = 0; t < 4; ++t) {
#pragma unroll
      for (int r = 0; r < 8; ++r)
        bounce[(r + mofs) * 64 + t * 16 + n] = lrelu(c2[t][r] + b2v[t]);
    }
    asm volatile("s_wait_dscnt 0" ::: "memory");

    // ---- layer 3: [16 x 64] @ [64 x 16] + b3 (+lrelu) (+resid) ----
    v8f c3 = v8f_zero();
#pragma unroll
    for (int kk = 0; kk < 16; ++kk) {
      const int k0 = kk * 4 + hi2;
      const int kp = k0 >> 1;
      const v2f a = *(const v2f*)(brow + k0);
      const v2f b = *(const v2f*)&s_w3[(kp << 5) + (n << 1)];
      c3 = wmma4(a, b, c3);
    }
#pragma unroll
    for (int r = 0; r < 8; ++r) {
      const int orow = r0 + r + mofs;
      float v = c3[r] + b3v;
      if (!FINAL_LIN) v = lrelu(v);
      if (RESID) v += resid[(size_t)orow * 16 + n];
      out[(size_t)orow * 16 + n] = v;
    }
  }
}

// ---------------------------------------------------------------------------
// Per-graph pooled stats: [count, mean(16), mad(16), max(16)] -> pooled[G,52]
// One 256-thread workgroup per (sorted, contiguous) graph segment.
// ---------------------------------------------------------------------------
__global__ void __launch_bounds__(256)
pool_kernel(const float* __restrict__ h, const int* __restrict__ starts,
            float* __restrict__ pooled)
{
  __shared__ float red[256 * 16];
  __shared__ float s_mean[16];
  const int g = blockIdx.x;
  const int tid = threadIdx.x;
  const int s = starts[g], e = starts[g + 1];
  const float cnt = (float)(e - s);
  const float inv = 1.f / fmaxf(cnt, 1.f);

  float acc[16], mx[16];
#pragma unroll
  for (int f = 0; f < 16; ++f) { acc[f] = 0.f; mx[f] = -__builtin_inff(); }
  for (int i = s + tid; i < e; i += 256) {
    const float* row = h + (size_t)i * 16;
#pragma unroll
    for (int f = 0; f < 16; ++f) {
      float v = row[f];
      acc[f] += v;
      mx[f] = fmaxf(mx[f], v);
    }
  }
  // sum -> mean
#pragma unroll
  for (int f = 0; f < 16; ++f) red[tid * 16 + f] = acc[f];
  __syncthreads();
  for (int off = 128; off > 0; off >>= 1) {
    if (tid < off)
      for (int f = 0; f < 16; ++f) red[tid * 16 + f] += red[(tid + off) * 16 + f];
    __syncthreads();
  }
  if (tid < 16) s_mean[tid] = red[tid] * inv;
  __syncthreads();
  // max
#pragma unroll
  for (int f = 0; f < 16; ++f) red[tid * 16 + f] = mx[f];
  __syncthreads();
  for (int off = 128; off > 0; off >>= 1) {
    if (tid < off)
      for (int f = 0; f < 16; ++f)
        red[tid * 16 + f] = fmaxf(red[tid * 16 + f], red[(tid + off) * 16 + f]);
    __syncthreads();
  }
  if (tid < 16) pooled[(size_t)g * 52 + 33 + tid] = red[tid];
  __syncthreads();
  // mad
#pragma unroll
  for (int f = 0; f < 16; ++f) acc[f] = 0.f;
  for (int i = s + tid; i < e; i += 256) {
    const float* row = h + (size_t)i * 16;
#pragma unroll
    for (int f = 0; f < 16; ++f) acc[f] += fabsf(row[f] - s_mean[f]);
  }
#pragma unroll
  for (int f = 0; f < 16; ++f) red[tid * 16 + f] = acc[f];
  __syncthreads();
  for (int off = 128; off > 0; off >>= 1) {
    if (tid < off)
      for (int f = 0; f < 16; ++f) red[tid * 16 + f] += red[(tid + off) * 16 + f];
    __syncthreads();
  }
  if (tid == 0) pooled[(size_t)g * 52] = cnt;
  if (tid < 16) {
    pooled[(size_t)g * 52 + 1 + tid]  = s_mean[tid];
    pooled[(size_t)g * 52 + 17 + tid] = red[tid] * inv;
  }
}

// ---------------------------------------------------------------------------
// Tiny per-graph MLP (glob / disc heads): w -> 64 -> 64 -> nout
// ---------------------------------------------------------------------------
__global__ void __launch_bounds__(64)
small_ffn_kernel(const float* __restrict__ in0, int stride0, int w0,
                 const float* __restrict__ in1, int w1,
                 const float* __restrict__ W1, const float* __restrict__ B1,
                 const float* __restrict__ W2, const float* __restrict__ B2,
                 const float* __restrict__ W3, const float* __restrict__ B3,
                 int nout, float* __restrict__ out, int ostride, int final_lin)
{
  __shared__ float s_in[64];
  __shared__ float s_h[64];
  const int g = blockIdx.x, t = threadIdx.x;
  if (t < w0) s_in[t] = in0[(size_t)g * stride0 + t];
  if (t < w1) s_in[w0 + t] = in1[(size_t)g * w1 + t];
  __syncthreads();
  const int w = w0 + w1;
  float a = B1[t];
  for (int k = 0; k < w; ++k) a += s_in[k] * W1[k * 64 + t];
  s_h[t] = lrelu(a);
  __syncthreads();
  float a2 = B2[t];
  for (int k = 0; k < 64; ++k) a2 += s_h[k] * W2[k * 64 + t];
  a2 = lrelu(a2);
  __syncthreads();
  s_h[t] = a2;
  __syncthreads();
  if (t < nout) {
    float o = B3[t];
    for (int k = 0; k < 64; ++k) o += s_h[k] * W3[k * nout + t];
    if (!final_lin) o = lrelu(o);
    out[(size_t)g * ostride + t] = o;
  }
}

// Segment starts via binary search over sorted batch (513 entries).
__global__ void starts_kernel(const int* __restrict__ batch, int n,
                              int* __restrict__ starts)
{
  int g = blockIdx.x * blockDim.x + threadIdx.x;
  if (g > NUM_GRAPHS) return;
  int lo = 0, hi = n;
  while (lo < hi) {
    int mid = (lo + hi) >> 1;
    if (batch[mid] < g) lo = mid + 1; else hi = mid;
  }
  starts[g] = lo;
}

extern "C" void kernel_launch(void* const* d_in, const int* in_sizes, int n_in,
                              void* d_out, int out_size, void* d_ws, size_t ws_size,
                              hipStream_t stream)
{
  (void)in_sizes; (void)n_in; (void)out_size; (void)ws_size;

  const float* x     = (const float*)d_in[0];
  const int*   batch = (const int*)  d_in[1];
  const float* cond  = (const float*)d_in[2];
  // params pytree: 'disc' (sorted before 'layers'), then 3 layers of
  // {'emb','glob','out'}, each FFN = (W,b) x3 -> 6 leaves.
  const float* dW1 = (const float*)d_in[3];
  const float* dB1 = (const float*)d_in[4];
  const float* dW2 = (const float*)d_in[5];
  const float* dB2 = (const float*)d_in[6];
  const float* dW3 = (const float*)d_in[7];
  const float* dB3 = (const float*)d_in[8];

  // workspace carve
  char* p = (char*)d_ws;
  auto carve = [&](size_t bytes) -> void* {
    void* r = (void*)p;
    p += (bytes + 255) & ~(size_t)255;
    return r;
  };
  float* X0     = (float*)carve((size_t)N_NODES * 16 * sizeof(float));
  float* X1     = (float*)carve((size_t)N_NODES * 16 * sizeof(float));
  float* H      = (float*)carve((size_t)N_NODES * 16 * sizeof(float));
  float* pooled = (float*)carve((size_t)NUM_GRAPHS * 52 * sizeof(float));
  float* xg     = (float*)carve((size_t)NUM_GRAPHS * 32 * sizeof(float));
  int*   starts = (int*)  carve(513 * sizeof(int));

  starts_kernel<<<3, 256, 0, stream>>>(batch, N_NODES, starts);

  const float* cur = x;
  float* bufs[2] = {X0, X1};
  for (int i = 0; i < 3; ++i) {
    const float* const* L = (const float* const*)(d_in + 9 + 18 * i);
    float* nxt = bufs[i & 1];
    // emb: 16 -> 64 -> 64 -> 16 (lrelu on all)
    node_ffn_kernel<16, false, false, false><<<1024, 128, 0, stream>>>(
        cur, nullptr, nullptr, nullptr,
        L[0], L[1], L[2], L[3], L[4], L[5], H);
    // pool h -> [count, mean, mad, max]
    pool_kernel<<<NUM_GRAPHS, 256, 0, stream>>>(H, starts, pooled);
    // glob: 49 -> 64 -> 64 -> 32 (lrelu on all)
    small_ffn_kernel<<<NUM_GRAPHS, 64, 0, stream>>>(
        pooled, 52, 49, nullptr, 0,
        L[6], L[7], L[8], L[9], L[10], L[11], 32, xg, 32, 0);
    // out: concat(h, xglob[batch]) 48 -> 64 -> 64 -> 16 (final linear) + resid
    node_ffn_kernel<48, true, true, true><<<1024, 128, 0, stream>>>(
        H, xg, batch, cur,
        L[12], L[13], L[14], L[15], L[16], L[17], nxt);
    cur = nxt;
  }
  // final pooling on x, then disc head: [pooled(49), cond(3)] -> 1
  pool_kernel<<<NUM_GRAPHS, 256, 0, stream>>>(cur, starts, pooled);
  small_ffn_kernel<<<NUM_GRAPHS, 64, 0, stream>>>(
      pooled, 52, 49, cond, 3,
      dW1, dB1, dW2, dB2, dW3, dB3, 1, (float*)d_out, 1, 1);
}